// LinearAdapter_90847148245154
// MI455X (gfx1250) — compile-verified
//
#include <hip/hip_runtime.h>

typedef float v2f __attribute__((ext_vector_type(2)));
typedef float v8f __attribute__((ext_vector_type(8)));

#define NUM_HEADS 16
#define D_HEAD    64
#define PHI_DIM   256
#define BATCH     4
#define MLEN      1024

#define TILE      128          // output tile edge (128x128 per workgroup)
#define LDS_STRIDE 68          // padded row stride (dwords) for 64-wide q/k tiles

__global__ __launch_bounds__(256)
void linear_adapter_gram_kernel(const float* __restrict__ phi,
                                const float* __restrict__ A,
                                const float* __restrict__ Bp,
                                float* __restrict__ out) {
    __shared__ float qs[TILE * LDS_STRIDE];
    __shared__ float ks[TILE * LDS_STRIDE];

    const int lane = threadIdx.x & 31;
    const int wave = threadIdx.x >> 5;     // 0..7
    const int l15  = lane & 15;
    const int hi   = lane >> 4;            // 0 or 1 (selects d+0 / d+2 half)

    const int bh = blockIdx.z;             // 0..63
    const int b  = bh >> 4;
    const int h  = bh & 15;
    const int i0 = blockIdx.y * TILE;      // output row block
    const int j0 = blockIdx.x * TILE;      // output col block

    // Per-lane row pointers for the A-operand (phi rows, row-major).
    const float* phiQ = phi + ((size_t)b * MLEN + i0 + wave * 16 + l15) * PHI_DIM;
    const float* phiK = phi + ((size_t)b * MLEN + j0 + wave * 16 + l15) * PHI_DIM;
    const float* Ah   = A  + (size_t)h * D_HEAD * PHI_DIM;
    const float* Bh   = Bp + (size_t)h * D_HEAD * PHI_DIM;

    const v8f vzero = {0.f, 0.f, 0.f, 0.f, 0.f, 0.f, 0.f, 0.f};

    // ---------------- Phase 1: q = phi_i @ A_h^T, k = phi_j @ B_h^T -------------
    // Each wave produces 16 rows x 64 cols of q (t=0) then of k (t=1).
    for (int t = 0; t < 2; ++t) {
        const float* prow = (t == 0) ? phiQ : phiK;   // A-operand rows
        const float* wmat = (t == 0) ? Ah   : Bh;     // weight rows (B-operand)
        float*       dst  = (t == 0) ? qs   : ks;

        v8f acc0 = vzero, acc1 = vzero, acc2 = vzero, acc3 = vzero;

        #pragma unroll 4
        for (int d = 0; d < PHI_DIM; d += 4) {
            // A-operand 16x4: lane l<16 -> row l, K={0,1}; lane>=16 -> K={2,3}
            v2f av = *(const v2f*)(prow + d + 2 * hi);
            // B-operand 4x16: B[kk][n] = W[n][d+kk]; per lane float2 along d
            v2f b0 = *(const v2f*)(wmat + (size_t)(0 * 16 + l15) * PHI_DIM + d + 2 * hi);
            v2f b1 = *(const v2f*)(wmat + (size_t)(1 * 16 + l15) * PHI_DIM + d + 2 * hi);
            v2f b2 = *(const v2f*)(wmat + (size_t)(2 * 16 + l15) * PHI_DIM + d + 2 * hi);
            v2f b3 = *(const v2f*)(wmat + (size_t)(3 * 16 + l15) * PHI_DIM + d + 2 * hi);

            acc0 = __builtin_amdgcn_wmma_f32_16x16x4_f32(false, av, false, b0, (short)0, acc0, false, false);
            acc1 = __builtin_amdgcn_wmma_f32_16x16x4_f32(false, av, false, b1, (short)0, acc1, false, false);
            acc2 = __builtin_amdgcn_wmma_f32_16x16x4_f32(false, av, false, b2, (short)0, acc2, false, false);
            acc3 = __builtin_amdgcn_wmma_f32_16x16x4_f32(false, av, false, b3, (short)0, acc3, false, false);
        }

        // Store C layout to LDS: VGPR r holds (row = r + 8*hi, col = nb*16 + l15)
        #pragma unroll
        for (int r = 0; r < 8; ++r) {
            const int row = wave * 16 + r + 8 * hi;
            dst[row * LDS_STRIDE + 0 * 16 + l15] = acc0[r];
            dst[row * LDS_STRIDE + 1 * 16 + l15] = acc1[r];
            dst[row * LDS_STRIDE + 2 * 16 + l15] = acc2[r];
            dst[row * LDS_STRIDE + 3 * 16 + l15] = acc3[r];
        }
    }

    __syncthreads();

    // ---------------- Phase 2: out_tile = q_tile @ k_tile^T (inner dim 64) -----
    v8f oacc[8];
    #pragma unroll
    for (int nb = 0; nb < 8; ++nb) oacc[nb] = vzero;

    #pragma unroll
    for (int d = 0; d < D_HEAD; d += 4) {
        // A-operand: q rows of this wave's 16-row strip
        v2f av = *(const v2f*)(&qs[(wave * 16 + l15) * LDS_STRIDE + d + 2 * hi]);
        #pragma unroll
        for (int nb = 0; nb < 8; ++nb) {
            // B-operand: B[kk][n] = k_tile[n][d+kk]
            v2f bv = *(const v2f*)(&ks[(nb * 16 + l15) * LDS_STRIDE + d + 2 * hi]);
            oacc[nb] = __builtin_amdgcn_wmma_f32_16x16x4_f32(false, av, false, bv, (short)0, oacc[nb], false, false);
        }
    }

    // ---------------- Write 16x128 strip of the output tile ---------------------
    float* op = out + (((size_t)bh) * MLEN + i0 + wave * 16) * MLEN + j0;
    #pragma unroll
    for (int nb = 0; nb < 8; ++nb) {
        #pragma unroll
        for (int r = 0; r < 8; ++r) {
            op[(size_t)(r + 8 * hi) * MLEN + nb * 16 + l15] = oacc[nb][r];
        }
    }
}

extern "C" void kernel_launch(void* const* d_in, const int* in_sizes, int n_in,
                              void* d_out, int out_size, void* d_ws, size_t ws_size,
                              hipStream_t stream) {
    const float* phi = (const float*)d_in[0];
    const float* A   = (const float*)d_in[1];
    const float* Bp  = (const float*)d_in[2];
    float* out       = (float*)d_out;

    dim3 grid(MLEN / TILE, MLEN / TILE, BATCH * NUM_HEADS);  // (8, 8, 64)
    dim3 block(256);
    linear_adapter_gram_kernel<<<grid, block, 0, stream>>>(phi, A, Bp, out);
}